// MMoEModel_49203145343596
// MI455X (gfx1250) — compile-verified
//
#include <hip/hip_runtime.h>
#include <math.h>

// MMoE on MI455X (gfx1250, wave32).
// d_ws layout: pooled[B*E] floats, then gate_value[B*NE] floats (~12.7 MB total).

#define B_   4096
#define S_   128
#define E_   768
#define HID_ 384
#define NH_  9
#define NE_  5

typedef float v2f __attribute__((ext_vector_type(2)));
typedef float v8f __attribute__((ext_vector_type(8)));

__device__ __forceinline__ float wave_reduce_add(float v) {
  // wave32 butterfly-free reduction: result valid in lane 0
  #pragma unroll
  for (int off = 16; off > 0; off >>= 1) v += __shfl_down(v, off, 32);
  return v;
}

// ---------------------------------------------------------------------------
// Kernel 1: masked-softmax attention pooling. One workgroup (8 waves) per b.
// Phase A: wave w computes scores for s = w*16 .. w*16+15 (coalesced dots).
// Phase B: exp (masked -> 0) + reduction in LDS.
// Phase C: pooled[e] = sum_s alpha_s * f[b,s,e]; re-read hits WGP$/L2.
// ---------------------------------------------------------------------------
__global__ __launch_bounds__(256)
void pool_kernel(const float* __restrict__ feature,
                 const int*   __restrict__ masks,
                 const float* __restrict__ attn_w,
                 const float* __restrict__ attn_b,
                 float*       __restrict__ pooled) {
  __shared__ float sc[S_];
  __shared__ float esc[S_];
  __shared__ float s_inv;

  const int b    = blockIdx.x;
  const int tid  = threadIdx.x;
  const int wave = tid >> 5;
  const int lane = tid & 31;
  const float* f = feature + (size_t)b * S_ * E_;
  const float ab = attn_b[0];

  // Phase A: scores
  #pragma unroll 1
  for (int i = 0; i < 16; ++i) {
    const int s = wave * 16 + i;
    const float* fr = f + (size_t)s * E_;
    float p = 0.f;
    #pragma unroll 4
    for (int e = lane; e < E_; e += 32) p += fr[e] * attn_w[e];
    p = wave_reduce_add(p);
    if (lane == 0) sc[s] = p + ab;
  }
  __syncthreads();

  // Phase B: masked exp + denom. exp(-1e9) underflows to 0 == reference mask.
  if (tid < S_) {
    const float v = (masks[b * S_ + tid] == 0) ? 0.f : __expf(sc[tid]);
    esc[tid] = v;
  }
  __syncthreads();
  if (wave == 0) {
    float v = esc[lane] + esc[lane + 32] + esc[lane + 64] + esc[lane + 96];
    v = wave_reduce_add(v);
    if (lane == 0) s_inv = 1.f / v;
  }
  __syncthreads();

  // Phase C: weighted sum. Thread t owns columns t, t+256, t+512.
  const float inv = s_inv;
  float a0 = 0.f, a1 = 0.f, a2 = 0.f;
  #pragma unroll 2
  for (int s = 0; s < S_; ++s) {
    const float al = esc[s] * inv;
    const float* fr = f + (size_t)s * E_;
    a0 = fmaf(al, fr[tid],        a0);
    a1 = fmaf(al, fr[tid + 256],  a1);
    a2 = fmaf(al, fr[tid + 512],  a2);
  }
  float* po = pooled + (size_t)b * E_;
  po[tid]       = a0;
  po[tid + 256] = a1;
  po[tid + 512] = a2;
}

// ---------------------------------------------------------------------------
// Kernel 2: category-selected gate MLP. One workgroup (384 threads) per b.
// Only the selected head h = category[b] is computed (reference computes all
// NH heads then selects; mathematically identical output).
// ---------------------------------------------------------------------------
__global__ __launch_bounds__(384)
void gate_kernel(const float* __restrict__ pooled,
                 const int*   __restrict__ category,
                 const float* __restrict__ gw1,
                 const float* __restrict__ gb1,
                 const float* __restrict__ gw2,
                 const float* __restrict__ gb2,
                 float*       __restrict__ gate_value) {
  __shared__ float pl[E_];
  __shared__ float gh[HID_];
  __shared__ float glog[NE_];

  const int b   = blockIdx.x;
  const int tid = threadIdx.x;
  const int h   = category[b];

  pl[tid]       = pooled[(size_t)b * E_ + tid];
  pl[tid + 384] = pooled[(size_t)b * E_ + tid + 384];
  __syncthreads();

  // gh[j] = relu(b1 + sum_e pooled[e] * W1[h][e][j]); thread j = tid.
  const float* w1 = gw1 + (size_t)h * E_ * HID_;
  float acc = gb1[h * HID_ + tid];
  #pragma unroll 4
  for (int e = 0; e < E_; ++e) acc = fmaf(pl[e], w1[(size_t)e * HID_ + tid], acc);
  gh[tid] = fmaxf(acc, 0.f);
  __syncthreads();

  const int wave = tid >> 5, lane = tid & 31;
  if (wave < NE_) {
    const float* w2 = gw2 + (size_t)h * HID_ * NE_;
    float p = 0.f;
    #pragma unroll 4
    for (int j = lane; j < HID_; j += 32) p = fmaf(gh[j], w2[j * NE_ + wave], p);
    p = wave_reduce_add(p);
    if (lane == 0) glog[wave] = p + gb2[h * NE_ + wave];
  }
  __syncthreads();

  if (tid == 0) {
    float m = glog[0];
    #pragma unroll
    for (int k = 1; k < NE_; ++k) m = fmaxf(m, glog[k]);
    float e[NE_], s = 0.f;
    #pragma unroll
    for (int k = 0; k < NE_; ++k) { e[k] = __expf(glog[k] - m); s += e[k]; }
    const float is = 1.f / s;
    #pragma unroll
    for (int k = 0; k < NE_; ++k) gate_value[b * NE_ + k] = e[k] * is;
  }
}

// ---------------------------------------------------------------------------
// Kernel 3: experts via V_WMMA_F32_16X16X4_F32, gate-mix into LDS rep tile,
// fused head + sigmoid. One workgroup (8 waves) per 16-row M-tile.
// N dimension = 5 experts * 384 = 1920 cols = 120 WMMA tiles; 15 per wave,
// processed 3 at a time sharing the A fragment.
// A frag layout (ISA 7.12.2): lane<16 -> M=lane, K={k,k+1}; lane>=16 -> K={k+2,k+3}.
// C tile: vgpr v, lane L -> row = v + 8*(L/16), col = L%16.
// ---------------------------------------------------------------------------
__global__ __launch_bounds__(256)
void expert_kernel(const float* __restrict__ pooled,
                   const float* __restrict__ expw,
                   const float* __restrict__ expb,
                   const float* __restrict__ gate_value,
                   const int*   __restrict__ category,
                   const float* __restrict__ head_w,
                   const float* __restrict__ head_b,
                   float*       __restrict__ out) {
  __shared__ float rep[16 * HID_];   // 24 KB mixed-expert accumulator
  __shared__ float gl[16 * NE_];

  const int tid  = threadIdx.x;
  const int wave = tid >> 5;
  const int lane = tid & 31;
  const int m0   = blockIdx.x * 16;

  for (int i = tid; i < 16 * HID_; i += 256) rep[i] = 0.f;
  if (tid < 16 * NE_)
    gl[tid] = gate_value[(size_t)(m0 + tid / NE_) * NE_ + (tid % NE_)];
  __syncthreads();

  const int half = lane >> 4;     // 0: K={k,k+1}, 1: K={k+2,k+3}
  const int l16  = lane & 15;
  const float* arow = pooled + (size_t)(m0 + l16) * E_;

  #pragma unroll 1
  for (int j0 = 0; j0 < 15; j0 += 3) {
    const int nt0 = wave * 15 + j0;
    const int nt1 = nt0 + 1;
    const int nt2 = nt0 + 2;
    const int col0 = nt0 * 16 + l16, ne0 = nt0 / 24, o0 = col0 - ne0 * HID_;
    const int col1 = nt1 * 16 + l16, ne1 = nt1 / 24, o1 = col1 - ne1 * HID_;
    const int col2 = nt2 * 16 + l16, ne2 = nt2 / 24, o2 = col2 - ne2 * HID_;
    const float* bp0 = expw + (size_t)ne0 * E_ * HID_ + o0;
    const float* bp1 = expw + (size_t)ne1 * E_ * HID_ + o1;
    const float* bp2 = expw + (size_t)ne2 * E_ * HID_ + o2;

    v8f c0 = {}, c1 = {}, c2 = {};
    #pragma unroll 2
    for (int k = 0; k < E_; k += 4) {
      const int kk = k + 2 * half;                  // even -> 8B aligned
      const v2f a = *(const v2f*)(arow + kk);       // A frag (shared x3)
      v2f b0; b0.x = bp0[(size_t)kk * HID_]; b0.y = bp0[(size_t)(kk + 1) * HID_];
      c0 = __builtin_amdgcn_wmma_f32_16x16x4_f32(false, a, false, b0,
                                                 (short)0, c0, false, false);
      v2f b1; b1.x = bp1[(size_t)kk * HID_]; b1.y = bp1[(size_t)(kk + 1) * HID_];
      c1 = __builtin_amdgcn_wmma_f32_16x16x4_f32(false, a, false, b1,
                                                 (short)0, c1, false, false);
      v2f b2; b2.x = bp2[(size_t)kk * HID_]; b2.y = bp2[(size_t)(kk + 1) * HID_];
      c2 = __builtin_amdgcn_wmma_f32_16x16x4_f32(false, a, false, b2,
                                                 (short)0, c2, false, false);
    }

    // bias + relu + gate-mix into LDS rep
    const float bi0 = expb[ne0 * HID_ + o0];
    const float bi1 = expb[ne1 * HID_ + o1];
    const float bi2 = expb[ne2 * HID_ + o2];
    #pragma unroll
    for (int v = 0; v < 8; ++v) {
      const int m = v + 8 * half;
      atomicAdd(&rep[m * HID_ + o0], gl[m * NE_ + ne0] * fmaxf(c0[v] + bi0, 0.f));
      atomicAdd(&rep[m * HID_ + o1], gl[m * NE_ + ne1] * fmaxf(c1[v] + bi1, 0.f));
      atomicAdd(&rep[m * HID_ + o2], gl[m * NE_ + ne2] * fmaxf(c2[v] + bi2, 0.f));
    }
  }
  __syncthreads();

  // fused head: wave handles rows 2*wave, 2*wave+1
  #pragma unroll
  for (int r = 0; r < 2; ++r) {
    const int m  = wave * 2 + r;
    const int bb = m0 + m;
    const int h  = category[bb];
    float p = 0.f;
    #pragma unroll 4
    for (int o = lane; o < HID_; o += 32)
      p = fmaf(rep[m * HID_ + o], head_w[h * HID_ + o], p);
    p = wave_reduce_add(p);
    if (lane == 0)
      out[bb] = 1.f / (1.f + __expf(-(p + head_b[h])));
  }
}

// ---------------------------------------------------------------------------
extern "C" void kernel_launch(void* const* d_in, const int* in_sizes, int n_in,
                              void* d_out, int out_size, void* d_ws, size_t ws_size,
                              hipStream_t stream) {
  (void)in_sizes; (void)n_in; (void)out_size; (void)ws_size;

  const float* feature  = (const float*)d_in[0];
  const int*   masks    = (const int*)  d_in[1];
  const int*   category = (const int*)  d_in[2];
  const float* attn_w   = (const float*)d_in[3];
  const float* attn_b   = (const float*)d_in[4];
  const float* gate_w1  = (const float*)d_in[5];
  const float* gate_b1  = (const float*)d_in[6];
  const float* gate_w2  = (const float*)d_in[7];
  const float* gate_b2  = (const float*)d_in[8];
  const float* expert_w = (const float*)d_in[9];
  const float* expert_b = (const float*)d_in[10];
  const float* head_w   = (const float*)d_in[11];
  const float* head_b   = (const float*)d_in[12];
  float* out = (float*)d_out;

  float* pooled     = (float*)d_ws;                 // B*E floats
  float* gate_value = pooled + (size_t)B_ * E_;     // B*NE floats

  pool_kernel<<<B_, 256, 0, stream>>>(feature, masks, attn_w, attn_b, pooled);
  gate_kernel<<<B_, 384, 0, stream>>>(pooled, category, gate_w1, gate_b1,
                                      gate_w2, gate_b2, gate_value);
  expert_kernel<<<B_ / 16, 256, 0, stream>>>(pooled, expert_w, expert_b,
                                             gate_value, category,
                                             head_w, head_b, out);
}